// MatchingPursuit_12146167513604
// MI455X (gfx1250) — compile-verified
//
#include <hip/hip_runtime.h>
#include <hip/hip_bf16.h>

typedef __attribute__((ext_vector_type(2))) float v2f;
typedef __attribute__((ext_vector_type(8))) float v8f;

#define BATCH 1024
#define IN_F  1024
#define OUT_F 2048

// ---- CDNA5 async global->LDS copy (GVS mode: SGPR base + per-lane u32 offset) ----
__device__ __forceinline__ void async_lds_b128(unsigned lds_addr, const float* sbase,
                                               unsigned byte_off) {
    asm volatile("global_load_async_to_lds_b128 %0, %1, %2"
                 :: "v"(lds_addr), "v"(byte_off), "s"(sbase)
                 : "memory");
}
__device__ __forceinline__ void wait_async0() {
    asm volatile("s_wait_asynccnt 0x0" ::: "memory");
}

// ---------------- Tiled f32 WMMA GEMM: C = A @ B ----------------
// BM=64, BN=128, BK=32. 256 threads = 8 wave32s; wave grid 2(M) x 4(N),
// each wave owns a 32x32 tile = 2x2 subtiles of 16x16 (v8f accumulators).
// Both A and B tiles are double-buffered in LDS and filled one K-block ahead
// with GLOBAL_LOAD_ASYNC_TO_LDS_B128 (ASYNCcnt); one barrier per K-block:
// each wave waits its own asyncs (s_wait_asynccnt 0) before the barrier.
// BT:  B is row-major (N x K), accessed as B[n,k] (C = A @ B^T); B staged
//      synchronously via register prefetch + transposed LDS stores.
// DUP: epilogue stores C to both C and C2 (used to seed beff = A_dot_b).
template<bool BT, bool DUP>
__global__ __launch_bounds__(256)
void gemm_wmma_f32(const float* __restrict__ A,
                   const float* __restrict__ B,
                   float* __restrict__ C,
                   float* __restrict__ C2,
                   int M, int N, int K)
{
    constexpr int BM = 64, BN = 128, BK = 32;
    __shared__ __align__(16) float As[2][BM][BK + 4];   // stride 36: conflict-free frags
    __shared__ __align__(16) float Bs[2][BK][BN];

    const int tid    = threadIdx.x;
    const int wave   = tid >> 5;
    const int lane   = tid & 31;
    const int lane16 = lane & 15;
    const int laneHi = lane >> 4;          // 0/1 -> K-pair select, M+8 select

    const int row0   = blockIdx.y * BM;
    const int col0   = blockIdx.x * BN;
    const int wmBase = (wave >> 2) * 32;   // 0 or 32
    const int wnBase = (wave & 3) * 32;    // 0,32,64,96

    float4 bR[4];                          // B regs (BT sync path only)

    auto issueA = [&](int kb, int nb) {    // 64x32 tile: 2 x b128 per thread, async
        #pragma unroll
        for (int i = 0; i < 2; ++i) {
            int idx = tid + i * 256, r = idx >> 3, c4 = (idx & 7) << 2;
            unsigned lds = (unsigned)(uintptr_t)&As[nb][r][c4];   // addr[31:0] = LDS offset
            unsigned off = (unsigned)(((size_t)(row0 + r) * K + kb + c4) * sizeof(float));
            async_lds_b128(lds, A, off);
        }
    };
    auto issueB = [&](int kb, int nb) {
        if (BT) {        // (N x K): prefetch regs; transpose-store at commit
            #pragma unroll
            for (int i = 0; i < 4; ++i) {
                int idx = tid + i * 256, nrow = idx >> 3, c4 = (idx & 7) << 2;
                bR[i] = *(const float4*)(B + (size_t)(col0 + nrow) * K + kb + c4);
            }
        } else {         // (K x N): 32x128 tile, 4 x b128 per thread, async
            #pragma unroll
            for (int i = 0; i < 4; ++i) {
                int idx = tid + i * 256, r = idx >> 5, c4 = (idx & 31) << 2;
                unsigned lds = (unsigned)(uintptr_t)&Bs[nb][r][c4];
                unsigned off = (unsigned)(((size_t)(kb + r) * N + col0 + c4) * sizeof(float));
                async_lds_b128(lds, B, off);
            }
        }
    };
    auto commitB_BT = [&](int nb) {        // BT only: transposed stores into LDS
        #pragma unroll
        for (int i = 0; i < 4; ++i) {
            int idx = tid + i * 256, nrow = idx >> 3, c4 = (idx & 7) << 2;
            Bs[nb][c4 + 0][nrow] = bR[i].x;
            Bs[nb][c4 + 1][nrow] = bR[i].y;
            Bs[nb][c4 + 2][nrow] = bR[i].z;
            Bs[nb][c4 + 3][nrow] = bR[i].w;
        }
    };

    v8f acc[2][2] = {};

    // ---- prologue: stage tile kb=0 into buffer 0 ----
    issueA(0, 0);
    issueB(0, 0);
    wait_async0();
    if (BT) commitB_BT(0);
    __syncthreads();

    int buf = 0;
    for (int kb = 0; kb < K; kb += BK, buf ^= 1) {
        const bool more = (kb + BK) < K;
        if (more) {                        // fill the other buffer while computing
            issueA(kb + BK, buf ^ 1);
            issueB(kb + BK, buf ^ 1);
        }

        // ---- compute: 8 K-microsteps of 4, 4 WMMAs each ----
        #pragma unroll
        for (int kk = 0; kk < BK; kk += 4) {
            const int kp = kk + (laneHi << 1);
            v2f af[2];
            #pragma unroll
            for (int tm = 0; tm < 2; ++tm) {
                int r = wmBase + tm * 16 + lane16;
                af[tm] = *(const v2f*)&As[buf][r][kp];
            }
            v2f bf[2];
            #pragma unroll
            for (int tn = 0; tn < 2; ++tn) {
                int c = wnBase + tn * 16 + lane16;
                bf[tn].x = Bs[buf][kp][c];
                bf[tn].y = Bs[buf][kp + 1][c];
            }
            #pragma unroll
            for (int tm = 0; tm < 2; ++tm)
                #pragma unroll
                for (int tn = 0; tn < 2; ++tn)
                    acc[tm][tn] = __builtin_amdgcn_wmma_f32_16x16x4_f32(
                        false, af[tm], false, bf[tn], (short)0, acc[tm][tn],
                        false, false);
        }

        if (more) {
            wait_async0();                 // own async writes into buf^1 landed
            if (BT) commitB_BT(buf ^ 1);
        }
        __syncthreads();                   // all reads of buf done + all fills visible
    }

    // ---- epilogue: D layout -> VGPR r holds M=r (lanes 0-15) / M=r+8 (16-31) ----
    #pragma unroll
    for (int tm = 0; tm < 2; ++tm) {
        #pragma unroll
        for (int tn = 0; tn < 2; ++tn) {
            int col = col0 + wnBase + tn * 16 + lane16;
            #pragma unroll
            for (int r = 0; r < 8; ++r) {
                int rowg = row0 + wmBase + tm * 16 + r + laneHi * 8;
                size_t o = (size_t)rowg * N + col;
                float val = acc[tm][tn][r];
                C[o] = val;
                if (DUP) C2[o] = val;
            }
        }
    }
}

// ---------------- zero-init ADMM state + encoded output ----------------
__global__ void init_state(float* __restrict__ v, float* __restrict__ u,
                           float* __restrict__ vsol, int* __restrict__ solved,
                           int total)
{
    int i = blockIdx.x * blockDim.x + threadIdx.x;
    if (i < total) { v[i] = 0.0f; u[i] = 0.0f; vsol[i] = 0.0f; }
    if (i < BATCH) solved[i] = 0;
}

// ------- one ADMM step: softshrink/dual update + beff update + row reduce -------
// beff = A_dot_b + v_new - u_new is maintained here so the GEMM's A operand is a
// single pure stream (async-copyable, no fused ALU). Frozen rows leave beff alone,
// matching the reference (v,u frozen => b_eff unchanged).
__global__ __launch_bounds__(256)
void admm_step(const float* __restrict__ xk, const float* __restrict__ Adotb,
               float* __restrict__ v, float* __restrict__ u,
               float* __restrict__ beff, float* __restrict__ v_sol,
               int* __restrict__ solved, float lambd, float tol)
{
    const int row = blockIdx.x;
    if (solved[row]) return;                  // frozen row: nothing observable changes

    const int N = OUT_F;
    const float* xr = xk    + (size_t)row * N;
    const float* ar = Adotb + (size_t)row * N;
    float* vr = v    + (size_t)row * N;
    float* ur = u    + (size_t)row * N;
    float* br = beff + (size_t)row * N;

    float vn[8];
    float dx2 = 0.0f, xn2 = 0.0f;
    #pragma unroll
    for (int j = 0; j < 8; ++j) {
        int c = threadIdx.x + j * 256;
        float xv = xr[c], uo = ur[c], vprev = vr[c], ad = ar[c];
        float z = xv + uo;
        float vnew = copysignf(fmaxf(fabsf(z) - lambd, 0.0f), z);  // softshrink
        float unew = uo + xv - vnew;
        vr[c] = vnew;
        ur[c] = unew;
        br[c] = ad + vnew - unew;            // next iteration's b_eff
        vn[j] = vnew;
        float d = vnew - vprev;
        dx2 += d * d;
        xn2 += vnew * vnew;
    }

    #pragma unroll
    for (int off = 16; off > 0; off >>= 1) {
        dx2 += __shfl_down(dx2, off);
        xn2 += __shfl_down(xn2, off);
    }
    __shared__ float sdx[8], sxn[8];
    __shared__ int sconv;
    const int lane = threadIdx.x & 31, wv = threadIdx.x >> 5;
    if (lane == 0) { sdx[wv] = dx2; sxn[wv] = xn2; }
    __syncthreads();
    if (threadIdx.x == 0) {
        float a = 0.0f, b = 0.0f;
        #pragma unroll
        for (int i = 0; i < 8; ++i) { a += sdx[i]; b += sxn[i]; }
        // dx/xn < tol  <=>  dx < tol*xn  (matches NaN-at-zero -> not converged)
        int conv = (sqrtf(a) < tol * sqrtf(b)) ? 1 : 0;
        sconv = conv;
        if (conv) solved[row] = 1;
    }
    __syncthreads();
    if (sconv) {
        float* sr = v_sol + (size_t)row * N;
        #pragma unroll
        for (int j = 0; j < 8; ++j) sr[threadIdx.x + j * 256] = vn[j];
    }
}

extern "C" void kernel_launch(void* const* d_in, const int* in_sizes, int n_in,
                              void* d_out, int out_size, void* d_ws, size_t ws_size,
                              hipStream_t stream)
{
    const float* x    = (const float*)d_in[0];   // (1024, 1024)
    const float* w    = (const float*)d_in[1];   // (2048, 1024)
    const float* Minv = (const float*)d_in[2];   // (2048, 2048)

    float* encoded = (float*)d_out;                       // (1024, 2048) == v_sol
    float* decoded = encoded + (size_t)BATCH * OUT_F;     // (1024, 1024)

    float* ws    = (float*)d_ws;
    float* Adotb = ws;
    float* beff  = Adotb + (size_t)BATCH * OUT_F;
    float* v     = beff  + (size_t)BATCH * OUT_F;
    float* u     = v     + (size_t)BATCH * OUT_F;
    float* xk    = u     + (size_t)BATCH * OUT_F;
    int*   solved = (int*)(xk + (size_t)BATCH * OUT_F);

    const int total = BATCH * OUT_F;
    init_state<<<(total + 255) / 256, 256, 0, stream>>>(v, u, encoded, solved, total);

    dim3 blk(256);
    // A_dot_b = x @ w^T : M=1024, N=2048, K=1024 (NT); dup-store seeds beff (v=u=0)
    gemm_wmma_f32<true, true><<<dim3(OUT_F / 128, BATCH / 64), blk, 0, stream>>>(
        x, w, Adotb, beff, BATCH, OUT_F, IN_F);

    for (int it = 0; it < 100; ++it) {
        // xk = beff @ M_inv : M=1024, N=2048, K=2048 (NN, both tiles async-to-LDS)
        gemm_wmma_f32<false, false><<<dim3(OUT_F / 128, BATCH / 64), blk, 0, stream>>>(
            beff, Minv, xk, nullptr, BATCH, OUT_F, OUT_F);
        admm_step<<<BATCH, 256, 0, stream>>>(xk, Adotb, v, u, beff, encoded, solved,
                                             0.2f, 1e-4f);
    }

    // decoded = v_sol @ w : M=1024, N=1024, K=2048 (NN; w is (K=2048, N=1024))
    gemm_wmma_f32<false, false><<<dim3(IN_F / 128, BATCH / 64), blk, 0, stream>>>(
        encoded, w, decoded, nullptr, BATCH, IN_F, OUT_F);
}